// TransformGNN_54735063220488
// MI455X (gfx1250) — compile-verified
//
#include <hip/hip_runtime.h>
#include <math.h>

// ---------------------------------------------------------------------------
// GNN: h1 = transform(x@w1+b1); agg = edge scatter-add; h2 = transform(...@w2+b2)
// pooled = [mean,max] per graph; out = sigmoid(pooled@w3+b3)
// Matmuls: V_WMMA_F32_16X16X4_F32 (exact fp32 on the CDNA5 matrix pipe).
// transform(): hardware V_TANH_F32 via __builtin_amdgcn_tanhf (gfx1250 trans op)
// instead of the ~40-VALU branchy OCML tanh expansion.
// Dominant cost: 5M-edge scatter; h/agg (64MB each) live in the 192MB L2.
// ---------------------------------------------------------------------------

typedef __attribute__((ext_vector_type(2))) float v2f;
typedef __attribute__((ext_vector_type(8))) float v8f;

#define HID 16

__device__ __forceinline__ float fast_tanh(float v) {
#if __has_builtin(__builtin_amdgcn_tanhf)
    return __builtin_amdgcn_tanhf(v);      // v_tanh_f32 (single TRANS32 op)
#else
    return tanhf(v);
#endif
}

__device__ __forceinline__ float xform(float v) {
    float y = fast_tanh(v);
    y = (y != y) ? 0.0f : y;                 // nan_to_num(nan=0)
    return fminf(10.0f, fmaxf(-10.0f, y));   // clip(-10, 10)
}

// float atomic max via monotone int/uint reinterpretation (init = -inf)
__device__ __forceinline__ void atomicMaxF(float* a, float v) {
    if (v >= 0.0f) atomicMax((int*)a, __float_as_int(v));
    else           atomicMin((unsigned int*)a, __float_as_uint(v));
}

// ---------------------------------------------------------------------------
__global__ void k_init(float* __restrict__ agg, float* __restrict__ deg,
                       float* __restrict__ hsum, float* __restrict__ hmax,
                       float* __restrict__ cnt, int n_nodes, int n_graphs) {
    long i = (long)blockIdx.x * blockDim.x + threadIdx.x;
    long total = (long)n_nodes * HID;
    long stride = (long)gridDim.x * blockDim.x;
    long gh = (long)n_graphs * HID;
    for (long idx = i; idx < total; idx += stride) {
        agg[idx] = 0.0f;
        if (idx < n_nodes) deg[idx] = 0.0f;
        if (idx < gh) { hsum[idx] = 0.0f; hmax[idx] = -INFINITY; }
        if (idx < n_graphs) cnt[idx] = 0.0f;
    }
}

// ---------------------------------------------------------------------------
// Layer 1: one wave = one 16-node tile. A = x tile (16x3 padded to 16x4),
// B = w1 (3x16 padded to 4x16), C seeded with bias. Single 16x16x4 f32 WMMA.
__global__ void k_layer1(const float* __restrict__ x, const float* __restrict__ w1,
                         const float* __restrict__ b1, float* __restrict__ h,
                         int n_tiles) {
    int lane = threadIdx.x & 31;
    int tile = blockIdx.x * (blockDim.x >> 5) + (threadIdx.x >> 5);
    if (tile >= n_tiles) return;           // wave-uniform: EXEC stays all-ones
    int half = lane >> 4;                  // 0: K={0,1}, 1: K={2,3}
    int lm   = lane & 15;                  // row (A) / column (B,C,D)
    long base = (long)tile * 16;
    long node = base + lm;

    v2f A, B;
    if (half == 0) {
        A.x = x[node * 3 + 0];  A.y = x[node * 3 + 1];
        B.x = w1[0 * 16 + lm];  B.y = w1[1 * 16 + lm];
    } else {
        A.x = x[node * 3 + 2];  A.y = 0.0f;    // K=3 zero pad
        B.x = w1[2 * 16 + lm];  B.y = 0.0f;
    }
    v8f acc;
    float bias = b1[lm];
    #pragma unroll
    for (int r = 0; r < 8; ++r) acc[r] = bias;   // C[m][n] = b1[n]

    acc = __builtin_amdgcn_wmma_f32_16x16x4_f32(false, A, false, B,
                                                (short)0, acc, false, false);

    #pragma unroll
    for (int r = 0; r < 8; ++r) {
        long m = base + half * 8 + r;
        h[m * HID + lm] = xform(acc[r]);   // lanes of a half store coalesced 64B
    }
}

// ---------------------------------------------------------------------------
// Edge scatter: 4 threads per edge; each moves one float4 of the source
// feature row and atomically adds it to the destination row. q==0 also
// accumulates the destination degree.
__global__ void k_scatter(const int* __restrict__ ei, const float* __restrict__ h,
                          float* __restrict__ agg, float* __restrict__ deg,
                          int n_edges) {
    long total = (long)n_edges * 4;
    long stride = (long)gridDim.x * blockDim.x;
    for (long t = (long)blockIdx.x * blockDim.x + threadIdx.x; t < total; t += stride) {
        long e = t >> 2;
        int  q = (int)(t & 3);
        int  r = ei[e];                 // edge_index[0][e] (dest)
        int  c = ei[(long)n_edges + e]; // edge_index[1][e] (src)
        float4 v = ((const float4*)(h + (long)c * HID))[q];
        float* dst = agg + (long)r * HID + q * 4;
        atomicAdd(dst + 0, v.x); atomicAdd(dst + 1, v.y);
        atomicAdd(dst + 2, v.z); atomicAdd(dst + 3, v.w);
        if (q == 0) atomicAdd(deg + r, 1.0f);
    }
}

// ---------------------------------------------------------------------------
// Layer 2 fused with pooling: A = (agg/deg) 16x16 tile, chained over 4 K-tiles
// of 16x16x4 WMMA; result transformed, then mean/max pooled via atomics.
__global__ void k_layer2(const float* __restrict__ agg, const float* __restrict__ deg,
                         const float* __restrict__ w2, const float* __restrict__ b2,
                         const int* __restrict__ batch,
                         float* __restrict__ hsum, float* __restrict__ hmax,
                         float* __restrict__ cnt, int n_tiles) {
    int lane = threadIdx.x & 31;
    int tile = blockIdx.x * (blockDim.x >> 5) + (threadIdx.x >> 5);
    if (tile >= n_tiles) return;           // wave-uniform exit
    int half = lane >> 4;
    int lm   = lane & 15;
    long base = (long)tile * 16;
    long node = base + lm;

    float inv = 1.0f / fmaxf(deg[node], 1.0f);   // per-row 1/deg (A rows)

    v8f acc;
    float bias = b2[lm];
    #pragma unroll
    for (int r = 0; r < 8; ++r) acc[r] = bias;

    #pragma unroll
    for (int kt = 0; kt < 4; ++kt) {
        int k0 = kt * 4 + half * 2;
        v2f A, B;
        A.x = agg[node * HID + k0 + 0] * inv;
        A.y = agg[node * HID + k0 + 1] * inv;
        B.x = w2[(k0 + 0) * 16 + lm];
        B.y = w2[(k0 + 1) * 16 + lm];
        acc = __builtin_amdgcn_wmma_f32_16x16x4_f32(false, A, false, B,
                                                    (short)0, acc, false, false);
    }

    #pragma unroll
    for (int r = 0; r < 8; ++r) {
        long m = base + half * 8 + r;
        float v = xform(acc[r]);
        int g = batch[m];
        atomicAdd(hsum + (long)g * HID + lm, v);
        atomicMaxF(hmax + (long)g * HID + lm, v);
        if (lm == 0) atomicAdd(cnt + g, 1.0f);   // lanes 0 & 16 cover all 16 rows
    }
}

// ---------------------------------------------------------------------------
__global__ void k_final(const float* __restrict__ hsum, const float* __restrict__ hmax,
                        const float* __restrict__ cnt, const float* __restrict__ w3,
                        const float* __restrict__ b3, float* __restrict__ out,
                        int n_graphs) {
    int g = blockIdx.x * blockDim.x + threadIdx.x;
    if (g >= n_graphs) return;
    float c = fmaxf(cnt[g], 1.0f);
    float acc = b3[0];
    #pragma unroll
    for (int f = 0; f < HID; ++f)
        acc += (hsum[(long)g * HID + f] / c) * w3[f];
    #pragma unroll
    for (int f = 0; f < HID; ++f) {
        float m = hmax[(long)g * HID + f];
        if (m == -INFINITY) m = 0.0f;            // empty-graph segment_max -> 0
        acc += m * w3[HID + f];
    }
    out[g] = 1.0f / (1.0f + expf(-acc));
}

// ---------------------------------------------------------------------------
extern "C" void kernel_launch(void* const* d_in, const int* in_sizes, int n_in,
                              void* d_out, int out_size, void* d_ws, size_t ws_size,
                              hipStream_t stream) {
    const float* x   = (const float*)d_in[0];
    const int*   ei  = (const int*)  d_in[1];
    const int*   bat = (const int*)  d_in[2];
    const float* w1  = (const float*)d_in[3];
    const float* b1  = (const float*)d_in[4];
    const float* w2  = (const float*)d_in[5];
    const float* b2  = (const float*)d_in[6];
    const float* w3  = (const float*)d_in[7];
    const float* b3  = (const float*)d_in[8];
    float* out = (float*)d_out;

    int n_nodes  = in_sizes[0] / 3;
    int n_edges  = in_sizes[1] / 2;
    int n_graphs = out_size;

    // workspace carve-up (~132.5 MB for the nominal sizes)
    float* h    = (float*)d_ws;                         // [N,16]
    float* agg  = h    + (size_t)n_nodes * HID;         // [N,16]
    float* deg  = agg  + (size_t)n_nodes * HID;         // [N]
    float* hsum = deg  + (size_t)n_nodes;               // [G,16]
    float* hmax = hsum + (size_t)n_graphs * HID;        // [G,16]
    float* cnt  = hmax + (size_t)n_graphs * HID;        // [G]

    const int T = 256;                                  // 8 wave32s per block

    long initN = (long)n_nodes * HID;
    int initBlocks = (int)((initN + T - 1) / T);
    k_init<<<initBlocks, T, 0, stream>>>(agg, deg, hsum, hmax, cnt, n_nodes, n_graphs);

    int n_tiles = n_nodes / 16;                         // 1M % 16 == 0
    int wavesPerBlock = T / 32;
    int tileBlocks = (n_tiles + wavesPerBlock - 1) / wavesPerBlock;
    k_layer1<<<tileBlocks, T, 0, stream>>>(x, w1, b1, h, n_tiles);

    long scatN = (long)n_edges * 4;
    int scatBlocks = (int)((scatN + T - 1) / T);
    k_scatter<<<scatBlocks, T, 0, stream>>>(ei, h, agg, deg, n_edges);

    k_layer2<<<tileBlocks, T, 0, stream>>>(agg, deg, w2, b2, bat,
                                           hsum, hmax, cnt, n_tiles);

    int finBlocks = (n_graphs + T - 1) / T;
    k_final<<<finBlocks, T, 0, stream>>>(hsum, hmax, cnt, w3, b3, out, n_graphs);
}